// GNN_49907519979972
// MI455X (gfx1250) — compile-verified
//
#include <hip/hip_runtime.h>

typedef float v2f __attribute__((ext_vector_type(2)));
typedef float v8f __attribute__((ext_vector_type(8)));

#define D 64
#define APITCH 68   // LDS pitch: A-fragment reads hit distinct banks (stride 68 floats per row)
#define ROWS 64     // nodes per block (4 row-tiles of 16; one per wave)

// ---------------- CSR build kernels ----------------

__global__ void k_zero(int* deg, int* fill, int n) {
    int i = blockIdx.x * blockDim.x + threadIdx.x;
    if (i < n) { deg[i] = 0; fill[i] = 0; }
}

__global__ void k_count(const int* dst, int* deg, int e) {
    int i = blockIdx.x * blockDim.x + threadIdx.x;
    if (i < e) atomicAdd(&deg[dst[i]], 1);
}

// single-block exclusive scan over n entries (n ~ 100K -> ~98 chunk iterations)
__global__ void k_scan(const int* deg, int* row_ptr, int n) {
    __shared__ int sm[1024];
    __shared__ int carry_s;
    int t = threadIdx.x;
    if (t == 0) carry_s = 0;
    __syncthreads();
    for (int base = 0; base < n; base += 1024) {
        int i = base + t;
        int d = (i < n) ? deg[i] : 0;
        sm[t] = d;
        __syncthreads();
        for (int off = 1; off < 1024; off <<= 1) {
            int v = (t >= off) ? sm[t - off] : 0;
            __syncthreads();
            sm[t] += v;
            __syncthreads();
        }
        int incl = sm[t];
        int carry = carry_s;          // read before update (barrier below orders it)
        if (i < n) row_ptr[i] = carry + incl - d;
        __syncthreads();
        if (t == 1023) carry_s = carry + incl;
        __syncthreads();
    }
    if (t == 0) row_ptr[n] = carry_s;
}

__global__ void k_fill(const int* srcv, const int* dstv, const int* row_ptr,
                       int* fill, int* col, int e) {
    int i = blockIdx.x * blockDim.x + threadIdx.x;
    if (i < e) {
        int d = dstv[i];
        int p = atomicAdd(&fill[d], 1);
        col[row_ptr[d] + p] = srcv[i];
    }
}

__global__ void k_invdeg(const int* deg, float* inv, int n) {
    int i = blockIdx.x * blockDim.x + threadIdx.x;
    if (i < n) inv[i] = 1.0f / fmaxf((float)deg[i], 1.0f);
}

// ---------------- fused SAGE layer: aggregate + (agg@Wl + bl + h@Wr) + ReLU ----------------
// grid: ceil(n/64) blocks of 128 threads (4 waves). Block owns 64 nodes; wave w owns rows
// [16w,16w+16). Weight staging amortized over 4 row-tiles; A-fragments held in registers
// and reused across all 4 output column-tiles (128 WMMAs per wave).

__global__ __launch_bounds__(128) void k_sage(
    const float* __restrict__ h_in, const int* __restrict__ row_ptr,
    const int* __restrict__ col, const float* __restrict__ inv_deg,
    const float* __restrict__ Wl, const float* __restrict__ bl,
    const float* __restrict__ Wr, float* __restrict__ h_out, int n)
{
    __shared__ float Wl_s[D * D];        // 16 KB
    __shared__ float Wr_s[D * D];        // 16 KB
    __shared__ float As[ROWS * APITCH];  // 17 KB: mean-aggregated neighbor rows
    __shared__ float bl_s[D];

    const int t = threadIdx.x;
    const int base = blockIdx.x * ROWS;

    __builtin_prefetch(Wl, 0, 3);        // global_prefetch_b8: weights -> cache early
    __builtin_prefetch(Wr, 0, 3);

    // stage weights in LDS
    {
        const float4* wl4 = (const float4*)Wl;
        const float4* wr4 = (const float4*)Wr;
        float4* wls4 = (float4*)Wl_s;
        float4* wrs4 = (float4*)Wr_s;
        for (int i = t; i < (D * D) / 4; i += 128) { wls4[i] = wl4[i]; wrs4[i] = wr4[i]; }
        if (t < D) bl_s[t] = bl[t];
    }

    // ---- aggregation: 8 threads per node row, 8 features each; 4 sub-tiles of 16 rows ----
    {
        const int rr = t >> 3;          // 0..15
        const int fg = t & 7;           // feature group (8 floats)
#pragma unroll
        for (int st = 0; st < 4; ++st) {
            const int r = st * 16 + rr;
            const int node = base + r;
            float* ap = &As[r * APITCH + fg * 8];
            if (node < n) {
                float acc[8];
#pragma unroll
                for (int i = 0; i < 8; ++i) acc[i] = 0.0f;
                const int beg = row_ptr[node];
                const int end = row_ptr[node + 1];
                for (int e2 = beg; e2 < end; ++e2) {
                    int s = col[e2];
                    const float4* sp = (const float4*)(h_in + (size_t)s * D + fg * 8);
                    float4 p0 = sp[0], p1 = sp[1];   // 25.6 MB feature table -> L2 hits
                    acc[0] += p0.x; acc[1] += p0.y; acc[2] += p0.z; acc[3] += p0.w;
                    acc[4] += p1.x; acc[5] += p1.y; acc[6] += p1.z; acc[7] += p1.w;
                }
                const float w = inv_deg[node];
#pragma unroll
                for (int i = 0; i < 8; ++i) ap[i] = acc[i] * w;
            } else {
#pragma unroll
                for (int i = 0; i < 8; ++i) ap[i] = 0.0f;
            }
        }
    }
    __syncthreads();

    // ---- WMMA phase ----
    const int lane = t & 31;
    const int wave = t >> 5;                 // row-tile id 0..3
    const int m    = lane & 15;
    const int koff = (lane >> 4) << 1;       // K offset: 0 (lanes 0-15) / 2 (lanes 16-31)
    const int rloc = wave * 16 + m;          // local A row 0..63
    const int rowg = base + rloc;            // global node for root-term fragments

    // load A-fragments once (agg from LDS, root features from global/L2); reuse across 4 n-tiles
    v2f aA[16], aH[16];
    const float* __restrict__ hrow = h_in + (size_t)((rowg < n) ? rowg : (n - 1)) * D;
#pragma unroll
    for (int k4 = 0; k4 < 16; ++k4) {
        const int kb = k4 * 4 + koff;
        aA[k4].x = As[rloc * APITCH + kb];
        aA[k4].y = As[rloc * APITCH + kb + 1];
        aH[k4].x = hrow[kb];
        aH[k4].y = hrow[kb + 1];
    }

    const int rb = (lane >> 4) << 3;         // C row base: 0 (lanes 0-15) / 8 (lanes 16-31)
#pragma unroll
    for (int nt = 0; nt < 4; ++nt) {
        const int nn = nt * 16 + m;          // output column
        v8f c = {0.f, 0.f, 0.f, 0.f, 0.f, 0.f, 0.f, 0.f};
#pragma unroll
        for (int k4 = 0; k4 < 16; ++k4) {    // agg @ Wl
            const int kb = k4 * 4 + koff;
            v2f b;
            b.x = Wl_s[kb * D + nn];
            b.y = Wl_s[(kb + 1) * D + nn];
            c = __builtin_amdgcn_wmma_f32_16x16x4_f32(false, aA[k4], false, b, (short)0, c, false, false);
        }
#pragma unroll
        for (int k4 = 0; k4 < 16; ++k4) {    // + h @ Wr
            const int kb = k4 * 4 + koff;
            v2f b;
            b.x = Wr_s[kb * D + nn];
            b.y = Wr_s[(kb + 1) * D + nn];
            c = __builtin_amdgcn_wmma_f32_16x16x4_f32(false, aH[k4], false, b, (short)0, c, false, false);
        }
        const float bias = bl_s[nn];
#pragma unroll
        for (int i = 0; i < 8; ++i) {        // C: VGPR i -> row i / i+8
            int node = base + wave * 16 + rb + i;
            if (node < n) h_out[(size_t)node * D + nn] = fmaxf(c[i] + bias, 0.0f);
        }
    }
}

// ---------------- global mean pool: one block per graph, batch sorted -> binary search ----------------

__global__ void k_pool(const float* __restrict__ h, const int* __restrict__ batch,
                       float* __restrict__ g, int n) {
    const int gid = blockIdx.x;
    const int t = threadIdx.x;  // feature
    int lo = 0, hi = n;
    while (lo < hi) { int mid = (lo + hi) >> 1; if (batch[mid] < gid) lo = mid + 1; else hi = mid; }
    const int beg = lo;
    lo = 0; hi = n;
    while (lo < hi) { int mid = (lo + hi) >> 1; if (batch[mid] < gid + 1) lo = mid + 1; else hi = mid; }
    const int end = lo;
    float acc = 0.0f;
    for (int i = beg; i < end; ++i) acc += h[(size_t)i * D + t];
    g[gid * D + t] = acc / fmaxf((float)(end - beg), 1.0f);
}

// ---------------- final MLP: relu(g@W0+b0) @ Wout + bout ----------------

__global__ void k_mlp(const float* __restrict__ g, const float* __restrict__ W0,
                      const float* __restrict__ b0, const float* __restrict__ Wo,
                      const float* __restrict__ bo, float* __restrict__ out, int outc) {
    __shared__ float trow[D];
    const int gid = blockIdx.x;
    const int t = threadIdx.x;
    float acc = b0[t];
    for (int k = 0; k < D; ++k) acc += g[gid * D + k] * W0[k * D + t];
    trow[t] = fmaxf(acc, 0.0f);
    __syncthreads();
    if (t < outc) {
        float o = bo[t];
        for (int k = 0; k < D; ++k) o += trow[k] * Wo[k * outc + t];
        out[gid * outc + t] = o;
    }
}

extern "C" void kernel_launch(void* const* d_in, const int* in_sizes, int n_in,
                              void* d_out, int out_size, void* d_ws, size_t ws_size,
                              hipStream_t stream) {
    const float* x     = (const float*)d_in[0];
    const int*   ei    = (const int*)d_in[1];
    const int*   batch = (const int*)d_in[2];
    const float* Wl[3] = {(const float*)d_in[3], (const float*)d_in[6], (const float*)d_in[9]};
    const float* blv[3]= {(const float*)d_in[4], (const float*)d_in[7], (const float*)d_in[10]};
    const float* Wr[3] = {(const float*)d_in[5], (const float*)d_in[8], (const float*)d_in[11]};
    const float* l0W   = (const float*)d_in[12];
    const float* l0b   = (const float*)d_in[13];
    const float* oW    = (const float*)d_in[14];
    const float* ob    = (const float*)d_in[15];
    float* out = (float*)d_out;

    const int n    = in_sizes[0] / D;
    const int e    = in_sizes[1] / 2;
    const int outc = in_sizes[15];          // out_b size == OUT
    const int g    = out_size / outc;       // number of graphs

    const int* srcv = ei;
    const int* dstv = ei + e;

    // workspace carve-out (~60 MB total)
    char* ws = (char*)d_ws;
    size_t off = 0;
    auto alloc = [&](size_t bytes) -> void* {
        void* p = (void*)(ws + off);
        off += (bytes + 255) & ~(size_t)255;
        return p;
    };
    int*   deg_i   = (int*)alloc((size_t)n * 4);
    int*   fill_i  = (int*)alloc((size_t)n * 4);
    int*   row_ptr = (int*)alloc(((size_t)n + 1) * 4);
    int*   colA    = (int*)alloc((size_t)e * 4);
    float* inv_d   = (float*)alloc((size_t)n * 4);
    float* hA      = (float*)alloc((size_t)n * D * 4);
    float* hB      = (float*)alloc((size_t)n * D * 4);
    float* gbuf    = (float*)alloc((size_t)g * D * 4);

    const int B256n = (n + 255) / 256;
    const int B256e = (e + 255) / 256;
    const int NT = (n + ROWS - 1) / ROWS;

    // CSR build (every call: deterministic, no caching)
    k_zero<<<B256n, 256, 0, stream>>>(deg_i, fill_i, n);
    k_count<<<B256e, 256, 0, stream>>>(dstv, deg_i, e);
    k_scan<<<1, 1024, 0, stream>>>(deg_i, row_ptr, n);
    k_fill<<<B256e, 256, 0, stream>>>(srcv, dstv, row_ptr, fill_i, colA, e);
    k_invdeg<<<B256n, 256, 0, stream>>>(deg_i, inv_d, n);

    // 3 fused SAGE layers (ping-pong buffers)
    k_sage<<<NT, 128, 0, stream>>>(x,  row_ptr, colA, inv_d, Wl[0], blv[0], Wr[0], hA, n);
    k_sage<<<NT, 128, 0, stream>>>(hA, row_ptr, colA, inv_d, Wl[1], blv[1], Wr[1], hB, n);
    k_sage<<<NT, 128, 0, stream>>>(hB, row_ptr, colA, inv_d, Wl[2], blv[2], Wr[2], hA, n);

    // pool + MLP head
    k_pool<<<g, D, 0, stream>>>(hA, batch, gbuf, n);
    k_mlp<<<g, D, 0, stream>>>(gbuf, l0W, l0b, oW, ob, out, outc);

    (void)n_in; (void)ws_size;
}